// GCNConv_66537633350263
// MI455X (gfx1250) — compile-verified
//
#include <hip/hip_runtime.h>

typedef __attribute__((ext_vector_type(2))) float v2f;
typedef __attribute__((ext_vector_type(8))) float v8f;

#define D 128          // D_IN == D_OUT == 128
#define TILE_M 64      // rows of X per block
#define KCHUNK 32      // K rows of W staged in LDS at a time
#define NMT (TILE_M / 16)

// ---------------------------------------------------------------------------
// Kernel 1: X' = X @ W using v_wmma_f32_16x16x4_f32.
// Block = 256 threads = 8 waves. Wave w owns N-tile w (cols 16w..16w+15).
// LDS: sX = 64x128 f32 stripe of X (32KB), sW = 32x128 f32 chunk of W (16KB).
// All 4 M-tiles are computed unconditionally (fully unrolled -> static VGPR
// accumulators, no v_movrel); only the stores are guarded for the tail block.
// ---------------------------------------------------------------------------
__global__ __launch_bounds__(256) void gcn_gemm_wmma(
    const float* __restrict__ X, const float* __restrict__ W,
    float* __restrict__ Xp, int n)
{
    __shared__ float sX[TILE_M * D];   // 32 KB
    __shared__ float sW[KCHUNK * D];   // 16 KB

    const int tid  = threadIdx.x;
    const int row0 = blockIdx.x * TILE_M;

    // cooperative load of the X stripe (float4, 16B aligned: D=128)
    for (int i = tid * 4; i < TILE_M * D; i += 256 * 4) {
        int r = row0 + (i >> 7);           // i / 128
        if (r < n) {
            *(float4*)&sX[i] = *(const float4*)&X[(size_t)r * D + (i & 127)];
        }
    }

    const int wave = tid >> 5;
    const int lane = tid & 31;
    const int n0   = wave * 16;            // this wave's output-column base
    const int mhi  = lane >> 4;            // half-wave select (0/1)
    const int mlo  = lane & 15;            // row within tile for A, col for B/C

    v8f acc0 = (v8f){};
    v8f acc1 = (v8f){};
    v8f acc2 = (v8f){};
    v8f acc3 = (v8f){};

    for (int kb = 0; kb < D; kb += KCHUNK) {
        __syncthreads();   // previous chunk fully consumed
        // stage W[kb..kb+31][0..127] into LDS
        for (int i = tid * 4; i < KCHUNK * D; i += 256 * 4) {
            *(float4*)&sW[i] = *(const float4*)&W[(size_t)(kb + (i >> 7)) * D + (i & 127)];
        }
        __syncthreads();   // chunk (and, first time, sX) visible

        const float* a0 = &sX[(0 * 16 + mlo) * D + kb + mhi * 2];
        const float* a1 = &sX[(1 * 16 + mlo) * D + kb + mhi * 2];
        const float* a2 = &sX[(2 * 16 + mlo) * D + kb + mhi * 2];
        const float* a3 = &sX[(3 * 16 + mlo) * D + kb + mhi * 2];
        const float* bp = &sW[(mhi * 2) * D + n0 + mlo];

        #pragma unroll
        for (int kk = 0; kk < KCHUNK; kk += 4) {
            // B 4x16 f32: VGPR0 holds K=kk(+2 for hi half), VGPR1 holds K=kk+1(+3)
            v2f b;
            b.x = bp[(kk + 0) * D];
            b.y = bp[(kk + 1) * D];
            // A 16x4 f32: lanes 0-15 -> K=kk,kk+1 ; lanes 16-31 -> K=kk+2,kk+3
            v2f av0 = *(const v2f*)(a0 + kk);
            v2f av1 = *(const v2f*)(a1 + kk);
            v2f av2 = *(const v2f*)(a2 + kk);
            v2f av3 = *(const v2f*)(a3 + kk);
            acc0 = __builtin_amdgcn_wmma_f32_16x16x4_f32(false, av0, false, b, (short)0, acc0, false, false);
            acc1 = __builtin_amdgcn_wmma_f32_16x16x4_f32(false, av1, false, b, (short)0, acc1, false, false);
            acc2 = __builtin_amdgcn_wmma_f32_16x16x4_f32(false, av2, false, b, (short)0, acc2, false, false);
            acc3 = __builtin_amdgcn_wmma_f32_16x16x4_f32(false, av3, false, b, (short)0, acc3, false, false);
        }
    }

    // C/D layout: VGPR r -> row (r + mhi*8) of the 16x16 tile, col = mlo.
    // Guard each M-tile store against the tail (n % 64 may be 32; always /16).
    v8f accs[NMT] = {acc0, acc1, acc2, acc3};
    #pragma unroll
    for (int mt = 0; mt < NMT; ++mt) {
        if (row0 + mt * 16 < n) {
            float* out = &Xp[(size_t)(row0 + mt * 16 + mhi * 8) * D + n0 + mlo];
            #pragma unroll
            for (int r = 0; r < 8; ++r) {
                out[(size_t)r * D] = accs[mt][r];
            }
        }
    }
}

// ---------------------------------------------------------------------------
// Kernel 2: CSR SpMM with symmetric GCN normalization.
// One wave32 per row; lane owns features [4*lane, 4*lane+4) as float4.
// out[i] = deg[i] * sum_e deg[col[e]] * Xp[col[e]]
// X' (51.2 MB) is L2-resident on MI455X (192 MB L2) -> gathers hit L2.
// ---------------------------------------------------------------------------
__global__ __launch_bounds__(256) void gcn_spmm_csr(
    const float* __restrict__ Xp, const float* __restrict__ deg,
    const long long* __restrict__ rp, const long long* __restrict__ ci,
    float* __restrict__ out, int n)
{
    const int row  = (blockIdx.x * blockDim.x + threadIdx.x) >> 5;
    const int lane = threadIdx.x & 31;
    if (row >= n) return;

    const long long e0 = rp[row];
    const long long e1 = rp[row + 1];

    float4 acc = {0.f, 0.f, 0.f, 0.f};
    for (long long e = e0; e < e1; ++e) {
        const long long c = ci[e];                 // uniform across the wave
        const float dc = deg[c];
        const float4 x = *(const float4*)&Xp[(size_t)c * D + lane * 4];
        acc.x += dc * x.x;
        acc.y += dc * x.y;
        acc.z += dc * x.z;
        acc.w += dc * x.w;
    }

    const float dr = deg[row];
    float4 r;
    r.x = acc.x * dr; r.y = acc.y * dr; r.z = acc.z * dr; r.w = acc.w * dr;
    *(float4*)&out[(size_t)row * D + lane * 4] = r;
}

// ---------------------------------------------------------------------------
extern "C" void kernel_launch(void* const* d_in, const int* in_sizes, int n_in,
                              void* d_out, int out_size, void* d_ws, size_t ws_size,
                              hipStream_t stream) {
    const float*     X   = (const float*)d_in[0];
    const float*     W   = (const float*)d_in[1];
    const float*     deg = (const float*)d_in[2];
    const long long* rp  = (const long long*)d_in[3];
    const long long* ci  = (const long long*)d_in[4];
    float*           out = (float*)d_out;

    const int n = in_sizes[2];           // N = number of nodes (degrees count)
    float* Xp = (float*)d_ws;            // X @ W scratch: N*128 floats

    dim3 g1((n + TILE_M - 1) / TILE_M);
    gcn_gemm_wmma<<<g1, 256, 0, stream>>>(X, W, Xp, n);

    dim3 g2((n + 7) / 8);                // 8 waves (rows) per 256-thread block
    gcn_spmm_csr<<<g2, 256, 0, stream>>>(Xp, deg, rp, ci, out, n);
}